// AlignedModule_41240275976458
// MI455X (gfx1250) — compile-verified
//
#include <hip/hip_runtime.h>
#include <math.h>

// MI455X / gfx1250, wave32. f32 WMMA 16x16x4 for the 1x1-conv GEMMs.
typedef __attribute__((ext_vector_type(2))) float v2f;
typedef __attribute__((ext_vector_type(8))) float v8f;

#define XS 65536  // x spatial = 16*64*64
#define YS 8192   // y spatial = 8*32*32

// ---------------------------------------------------------------------------
// 1) pooled_x[n*64+c] = mean over spatial of x  (mean commutes with 1x1 conv)
// ---------------------------------------------------------------------------
__global__ void pool_kernel(const float* __restrict__ x, float* __restrict__ pooled) {
  __shared__ float red[256];
  int t = threadIdx.x;
  const float* px = x + (size_t)blockIdx.x * XS;
  float s = 0.f;
  for (int i = t; i < XS; i += 256) s += px[i];
  red[t] = s;
  __syncthreads();
  for (int k = 128; k > 0; k >>= 1) {
    if (t < k) red[t] += red[t + k];
    __syncthreads();
  }
  if (t == 0) pooled[blockIdx.x] = red[0] * (1.0f / (float)XS);
}

// ---------------------------------------------------------------------------
// 2) Fold the whole low branch into one 64x64 matrix per batch:
//    pooled_low = w_down_l @ pooled_x
//    atten      = sigmoid(w_att_atten @ pooled_low)
//    W_low[n]   = (w_att_conv * (1+atten[n]) per-column) @ w_down_l
// ---------------------------------------------------------------------------
__global__ void weights_kernel(const float* __restrict__ pooled,
                               const float* __restrict__ w_down_l,
                               const float* __restrict__ w_att_atten,
                               const float* __restrict__ w_att_conv,
                               float* __restrict__ W_low) {
  __shared__ float pl[128];
  __shared__ float sc[128];
  int t = threadIdx.x;
  if (t < 128) {
    int n = t >> 6, c = t & 63;
    float s = 0.f;
    for (int i = 0; i < 64; i++) s += w_down_l[c * 64 + i] * pooled[n * 64 + i];
    pl[t] = s;
  }
  __syncthreads();
  if (t < 128) {
    int n = t >> 6, c = t & 63;
    float s = 0.f;
    for (int i = 0; i < 64; i++) s += w_att_atten[c * 64 + i] * pl[n * 64 + i];
    sc[t] = 1.f + 1.f / (1.f + expf(-s));
  }
  __syncthreads();
  for (int e = t; e < 2 * 64 * 64; e += 256) {
    int n = e >> 12, o = (e >> 6) & 63, j = e & 63;
    float s = 0.f;
    for (int c = 0; c < 64; c++)
      s += w_att_conv[o * 64 + c] * sc[n * 64 + c] * w_down_l[c * 64 + j];
    W_low[e] = s;  // [n][o][j], row-major 64x64 per batch
  }
}

// ---------------------------------------------------------------------------
// 3) 1x1 conv as GEMM on the matrix pipe: Out[64,S] = W[64xC] @ X[C,S].
//    One wave computes a 64(M) x 16(N) strip with V_WMMA_F32_16X16X4_F32.
//    f32 A (16x4) / B (4x16) fragments: float2/lane; lanes 0-15 K={0,1},
//    lanes 16-31 K={2,3}. C/D: v8f, VGPR r -> row m0+r (lanes 0-15) or
//    m0+8+r (lanes 16-31), column = lane&15.
// ---------------------------------------------------------------------------
__global__ void conv1x1_wmma(const float* __restrict__ X,   // [N][C][S]
                             const float* __restrict__ Wm,  // [.][64][C]
                             float* __restrict__ Out,       // [N][out_ctotal][S]
                             int C, int S, int w_nstride,
                             int out_ctotal, int out_off) {
  int wave = (int)((blockIdx.x * blockDim.x + threadIdx.x) >> 5);
  int lane = threadIdx.x & 31;
  int strips = S >> 4;
  int n = wave / strips;                 // wave-uniform
  int s0 = (wave - n * strips) << 4;
  int col = lane & 15;                   // N column / M row within half
  int krow = (lane >> 4) << 1;           // 0 or 2
  const float* Xb = X + (size_t)n * C * S + (size_t)s0 + col;
  const float* Wb = Wm + (size_t)n * w_nstride;
  v8f acc[4] = {};
  for (int k = 0; k < C; k += 4) {
    v2f b;
    b.x = Xb[(size_t)(k + krow) * S];
    b.y = Xb[(size_t)(k + krow + 1) * S];
#pragma unroll
    for (int tt = 0; tt < 4; tt++) {     // 4 M-tiles -> 64 out channels
      const float* wp = Wb + (size_t)(tt * 16 + col) * C + (k + krow);
      v2f a;
      a.x = wp[0];
      a.y = wp[1];
      acc[tt] = __builtin_amdgcn_wmma_f32_16x16x4_f32(
          false, a, false, b, (short)0, acc[tt], false, false);
    }
  }
  int nhalf = (lane >> 4) * 8;
  float* Ob = Out + ((size_t)n * out_ctotal + out_off) * S + s0 + col;
#pragma unroll
  for (int tt = 0; tt < 4; tt++) {
    int mbase = tt * 16 + nhalf;
#pragma unroll
    for (int r = 0; r < 8; r++) Ob[(size_t)(mbase + r) * S] = acc[tt][r];
  }
}

// ---------------------------------------------------------------------------
// 4) Trilinear upsample (align_corners=True) of hf [2,64,8,32,32] into
//    concat channels 0..63 at [2,128,16,64,64]. Separable weights folded
//    into 8 taps; coalesced over w across the wave.
// ---------------------------------------------------------------------------
__global__ void upsample_kernel(const float* __restrict__ hf, float* __restrict__ cc) {
  int v = blockIdx.x * 256 + threadIdx.x;
  int n = v >> 16;
  int sp = v & 65535;
  int d = sp >> 12, h = (sp >> 6) & 63, w = sp & 63;
  float pd = d * (7.f / 15.f);
  float ph = h * (31.f / 63.f);
  float pw = w * (31.f / 63.f);
  int d0 = min((int)pd, 7);  int d1 = min(d0 + 1, 7);  float fd = pd - (float)d0;
  int h0 = min((int)ph, 31); int h1 = min(h0 + 1, 31); float fh = ph - (float)h0;
  int w0 = min((int)pw, 31); int w1 = min(w0 + 1, 31); float fw = pw - (float)w0;
  int o000 = (d0 * 32 + h0) * 32 + w0, o001 = (d0 * 32 + h0) * 32 + w1;
  int o010 = (d0 * 32 + h1) * 32 + w0, o011 = (d0 * 32 + h1) * 32 + w1;
  int o100 = (d1 * 32 + h0) * 32 + w0, o101 = (d1 * 32 + h0) * 32 + w1;
  int o110 = (d1 * 32 + h1) * 32 + w0, o111 = (d1 * 32 + h1) * 32 + w1;
  float gd = 1.f - fd, gh = 1.f - fh, gw = 1.f - fw;
  float c000 = gd * gh * gw, c001 = gd * gh * fw;
  float c010 = gd * fh * gw, c011 = gd * fh * fw;
  float c100 = fd * gh * gw, c101 = fd * gh * fw;
  float c110 = fd * fh * gw, c111 = fd * fh * fw;
  const float* Hn = hf + (size_t)n * 64 * YS;
  float* Cn = cc + (size_t)n * 128 * XS + sp;
  for (int c = 0; c < 64; c++) {
    const float* Hc = Hn + (size_t)c * YS;
    float s = Hc[o000] * c000 + Hc[o001] * c001 + Hc[o010] * c010 + Hc[o011] * c011 +
              Hc[o100] * c100 + Hc[o101] * c101 + Hc[o110] * c110 + Hc[o111] * c111;
    Cn[(size_t)c * XS] = s;
  }
}

// ---------------------------------------------------------------------------
// 5) Fused: 3x3x3 conv (128ch -> 3 flow values, weights staged in LDS,
//    broadcast reads) + grid construction + trilinear grid_sample of y
//    (align_corners=False, zeros padding). flow is single-use -> never
//    materialized in HBM.
// ---------------------------------------------------------------------------
__global__ void flow_sample_kernel(const float* __restrict__ cc,
                                   const float* __restrict__ y,
                                   const float* __restrict__ wf,  // [3][128][27]
                                   float* __restrict__ out) {
  __shared__ float wl[3 * 128 * 27];  // 41472 B
  int t = threadIdx.x;
  for (int i = t; i < 3 * 128 * 27; i += 256) wl[i] = wf[i];
  __syncthreads();
  int v = blockIdx.x * 256 + t;
  int n = v >> 16;
  int sp = v & 65535;
  int d = sp >> 12, h = (sp >> 6) & 63, w = sp & 63;

  int offs[27];
  unsigned mask = 0;
#pragma unroll
  for (int kd = 0; kd < 3; kd++)
#pragma unroll
    for (int kh = 0; kh < 3; kh++)
#pragma unroll
      for (int kw = 0; kw < 3; kw++) {
        int tap = (kd * 3 + kh) * 3 + kw;
        int dd = d + kd - 1, hh = h + kh - 1, ww = w + kw - 1;
        bool ok = (dd >= 0 && dd < 16 && hh >= 0 && hh < 64 && ww >= 0 && ww < 64);
        offs[tap] = ok ? ((dd << 12) + (hh << 6) + ww) : 0;
        if (ok) mask |= (1u << tap);
      }

  const float* Xn = cc + (size_t)n * 128 * XS;
  float f0 = 0.f, f1 = 0.f, f2 = 0.f;
  for (int c = 0; c < 128; c++) {
    const float* Xc = Xn + (size_t)c * XS;
    const float* wc = wl + c * 27;
#pragma unroll
    for (int tap = 0; tap < 27; tap++) {
      float xv = ((mask >> tap) & 1u) ? Xc[offs[tap]] : 0.f;
      f0 = fmaf(xv, wc[tap], f0);
      f1 = fmaf(xv, wc[3456 + tap], f1);
      f2 = fmaf(xv, wc[6912 + tap], f2);
    }
  }

  // vgrid = (coord + flow) / (W,H,D of x); then torch-style unnormalize on y dims
  float gx = ((float)w + f0) * (1.f / 64.f);
  float gy = ((float)h + f1) * (1.f / 64.f);
  float gz = ((float)d + f2) * (1.f / 16.f);
  float ix = ((gx + 1.f) * 32.f - 1.f) * 0.5f;
  float iy = ((gy + 1.f) * 32.f - 1.f) * 0.5f;
  float iz = ((gz + 1.f) * 8.f - 1.f) * 0.5f;
  int x0 = (int)floorf(ix), y0 = (int)floorf(iy), z0 = (int)floorf(iz);
  float fx = ix - (float)x0, fy = iy - (float)y0, fz = iz - (float)z0;

  int idx8[8];
  float wg8[8];
#pragma unroll
  for (int dz = 0; dz < 2; dz++)
#pragma unroll
    for (int dy = 0; dy < 2; dy++)
#pragma unroll
      for (int dx = 0; dx < 2; dx++) {
        int zc = z0 + dz, yc = y0 + dy, xc = x0 + dx;
        bool ok = (zc >= 0 && zc < 8 && yc >= 0 && yc < 32 && xc >= 0 && xc < 32);
        int zl = min(max(zc, 0), 7), yl = min(max(yc, 0), 31), xl = min(max(xc, 0), 31);
        int k = (dz * 2 + dy) * 2 + dx;
        idx8[k] = (zl * 32 + yl) * 32 + xl;
        float wz = dz ? fz : 1.f - fz;
        float wy = dy ? fy : 1.f - fy;
        float wx = dx ? fx : 1.f - fx;
        wg8[k] = ok ? wz * wy * wx : 0.f;
      }

  const float* Yn = y + (size_t)n * 128 * YS;
  float* On = out + (size_t)n * 128 * XS + sp;
  for (int c = 0; c < 128; c++) {
    const float* Yc = Yn + (size_t)c * YS;
    float s = 0.f;
#pragma unroll
    for (int k = 0; k < 8; k++) s = fmaf(Yc[idx8[k]], wg8[k], s);
    On[(size_t)c * XS] = s;
  }
}

// ---------------------------------------------------------------------------
extern "C" void kernel_launch(void* const* d_in, const int* in_sizes, int n_in,
                              void* d_out, int out_size, void* d_ws, size_t ws_size,
                              hipStream_t stream) {
  (void)in_sizes; (void)n_in; (void)out_size; (void)ws_size;
  const float* x           = (const float*)d_in[0];  // [2,64,16,64,64]
  const float* y           = (const float*)d_in[1];  // [2,128,8,32,32]
  const float* w_down_h    = (const float*)d_in[2];  // [64,128]
  const float* w_down_l    = (const float*)d_in[3];  // [64,64]
  const float* w_flow      = (const float*)d_in[4];  // [3,128,3,3,3]
  const float* w_att_atten = (const float*)d_in[5];  // [64,64]
  const float* w_att_conv  = (const float*)d_in[6];  // [64,64]
  float* out = (float*)d_out;                        // [2,128,16,64,64]

  float* ws     = (float*)d_ws;
  float* pooled = ws;                       // 128
  float* W_low  = ws + 128;                 // 2*64*64 = 8192
  float* hf_lo  = W_low + 8192;             // 2*64*YS = 1,048,576
  float* concat = hf_lo + 2 * 64 * YS;      // 2*128*XS = 16,777,216 (~64 MB)

  // low-branch fold
  pool_kernel<<<128, 256, 0, stream>>>(x, pooled);
  weights_kernel<<<1, 256, 0, stream>>>(pooled, w_down_l, w_att_atten, w_att_conv, W_low);

  // low = W_low[n] @ x  -> concat channels 64..127   (8192 waves, 4/block)
  conv1x1_wmma<<<2048, 128, 0, stream>>>(x, W_low, concat, 64, XS, 64 * 64, 128, 64);
  // hf_lo = w_down_h @ y (low-res)                   (1024 waves, 4/block)
  conv1x1_wmma<<<256, 128, 0, stream>>>(y, w_down_h, hf_lo, 128, YS, 0, 64, 0);

  // upsample hf into concat channels 0..63
  upsample_kernel<<<512, 256, 0, stream>>>(hf_lo, concat);
  // fused conv3x3 -> flow -> grid_sample(y) -> d_out
  flow_sample_kernel<<<512, 256, 0, stream>>>(concat, y, w_flow, out);
}